// Block_76759655514736
// MI455X (gfx1250) — compile-verified
//
#include <hip/hip_runtime.h>
#include <hip/hip_bf16.h>
#include <math.h>

// ---------------------------------------------------------------------------
// Transformer block forward for MI455X (gfx1250).
// fp32 WMMA (16x16x4) + Tensor Data Mover staging of all LDS tiles.
// B=64, T=256, C=384, H=6, D=64, M=B*T=16384.
// ---------------------------------------------------------------------------

typedef float        v2f __attribute__((ext_vector_type(2)));
typedef float        v8f __attribute__((ext_vector_type(8)));
typedef unsigned int v4u __attribute__((ext_vector_type(4)));
typedef int          v8i __attribute__((ext_vector_type(8)));
typedef int          v4i __attribute__((ext_vector_type(4)));

#define EMBED     384
#define SEQ       256
#define BATCH     64
#define NHEAD     6
#define HEADD     64
#define MROWS     (BATCH * SEQ)       // 16384
#define QKPITCH   68   // row-major b64 frag reads: pitch%64==4 -> conflict-free
#define VPITCH    72   // col-major b32 frag reads: 2*pitch%64==16 -> conflict-free
#define BPITCH    72   // GEMM weight panel, col-major reads

__device__ __forceinline__ v8f wmma_f32(v2f a, v2f b, v8f c) {
  // (neg_a, A, neg_b, B, c_mod, C, reuse_a, reuse_b)
  return __builtin_amdgcn_wmma_f32_16x16x4_f32(false, a, false, b, (short)0, c,
                                               false, false);
}

// TDM 2D tile load: `rows` x 64 fp32 tile, row stride `stride_elems`,
// LDS padded by (pad_code+1) dwords every 64 dwords -> pitch 64+pad_code+1.
__device__ __forceinline__ void tdm_load_2d(const float* gsrc, unsigned lds_byte,
                                            unsigned rows, unsigned stride_elems,
                                            unsigned pad_code) {
  const unsigned long long ga = (unsigned long long)(uintptr_t)gsrc;
  v4u g0;
  g0.x = 1u;                                          // count=1, user mode
  g0.y = lds_byte;                                    // lds_addr
  g0.z = (unsigned)(ga & 0xffffffffu);                // global_addr[31:0]
  g0.w = (unsigned)((ga >> 32) & 0x01ffffffu) | (2u << 30);  // addr[56:32]|type=2
  v8i g1;
  g1[0] = (int)((2u << 16)            // data_size = 4B
              | (1u << 20)            // pad_enable
              | (5u << 22)            // pad_interval code 5 -> every 64 dwords
              | (pad_code << 25));    // pad_amount code -> pad_code+1 dwords
  g1[1] = (int)(64u << 16);           // tensor_dim0 = 64
  g1[2] = (int)(rows << 16);          // tensor_dim0 hi=0 | tensor_dim1 = rows
  g1[3] = (int)(64u << 16);           // tensor_dim1 hi=0 | tile_dim0 = 64
  g1[4] = (int)rows;                  // tile_dim1 = rows, tile_dim2 = 0
  g1[5] = (int)stride_elems;          // tensor_dim0_stride (elements)
  g1[6] = 0;
  g1[7] = 0;
  const v4i z4 = {0, 0, 0, 0};
  const v8i z8 = {0, 0, 0, 0, 0, 0, 0, 0};
  __builtin_amdgcn_tensor_load_to_lds(g0, g1, z4, z4, z8, 0);
}

// ---------------------------------------------------------------------------
// LayerNorm: one wave per row of 384. 8 rows per 256-thread block.
// ---------------------------------------------------------------------------
__global__ void ln_kernel(const float* __restrict__ x,
                          const float* __restrict__ g,
                          const float* __restrict__ b,
                          float* __restrict__ out) {
  const int wave = threadIdx.x >> 5;
  const int lane = threadIdx.x & 31;
  const int row  = blockIdx.x * 8 + wave;
  const float* xr = x + (size_t)row * EMBED;

  float v[12];
  float s = 0.f;
#pragma unroll
  for (int i = 0; i < 12; ++i) { v[i] = xr[lane + 32 * i]; s += v[i]; }
#pragma unroll
  for (int off = 16; off > 0; off >>= 1) s += __shfl_xor(s, off, 32);
  const float mu = s * (1.f / EMBED);

  float var = 0.f;
#pragma unroll
  for (int i = 0; i < 12; ++i) { const float d = v[i] - mu; var += d * d; }
#pragma unroll
  for (int off = 16; off > 0; off >>= 1) var += __shfl_xor(var, off, 32);
  var *= (1.f / EMBED);
  const float rstd = rsqrtf(var + 1e-5f);

  float* orow = out + (size_t)row * EMBED;
#pragma unroll
  for (int i = 0; i < 12; ++i) {
    const int c = lane + 32 * i;
    orow[c] = (v[i] - mu) * rstd * g[c] + b[c];
  }
}

// ---------------------------------------------------------------------------
// WMMA GEMM: out[16384,384] = A[16384,384] x W + bias (+res)(relu).
// Weight element for column `col`: W[(col>>6)*BLOCKSTRIDE + k*LDB + (col&63)]
// The 64-wide N-block tile (64 x 384, stride LDB) is DMA'd into LDS by the
// TDM (pitch 72). Block tile 128(M) x 64(N); 8 waves in a 4x2 grid, each wave
// owns a 32x32 register tile -> 4 WMMAs per k-step from 2 global b64 (A) +
// 4 LDS b32 (B, pairable to ds_load_2addr) loads.
// ---------------------------------------------------------------------------
template <int BLOCKSTRIDE, int LDB, bool RELU, bool RES>
__global__ void gemm_kernel(const float* __restrict__ A,
                            const float* __restrict__ W,
                            const float* __restrict__ bias,
                            const float* __restrict__ R,
                            float* __restrict__ out) {
  constexpr int K = EMBED;
  extern __shared__ float Bs[];            // [384][BPITCH] = 110.6 KB

  const int tid  = threadIdx.x;
  const int wave = tid >> 5;
  const int lane = tid & 31;
  const int col0 = blockIdx.x * 64;
  const int mb   = blockIdx.y * 128;

  if (wave == 0) {                         // TDM: stage 64x384 weight panel
    tdm_load_2d(W + (size_t)(col0 >> 6) * BLOCKSTRIDE,
                (unsigned)(uintptr_t)Bs, K, LDB, 7 /*pad 8 -> pitch 72*/);
    __builtin_amdgcn_s_wait_tensorcnt(0);
  }
  __syncthreads();

  const int m    = lane & 15;
  const int half = lane >> 4;
  const int wm   = wave & 3;               // 0..3 -> M offset
  const int wn   = wave >> 2;              // 0..1 -> N offset
  const int row0 = mb + wm * 32 + m;
  const float* arow0 = A + (size_t)row0 * EMBED + 2 * half;
  const float* arow1 = arow0 + (size_t)16 * EMBED;
  const float* bcol0 = Bs + (2 * half) * BPITCH + wn * 32 + m;   // m acts as n
  const float* bcol1 = bcol0 + 16;

  v8f c00 = {}, c01 = {}, c10 = {}, c11 = {};
#pragma unroll 2
  for (int kb = 0; kb < K; kb += 4) {
    const v2f a0 = *(const v2f*)(arow0 + kb);
    const v2f a1 = *(const v2f*)(arow1 + kb);
    v2f b0, b1;
    b0.x = bcol0[kb * BPITCH];
    b0.y = bcol0[(kb + 1) * BPITCH];
    b1.x = bcol1[kb * BPITCH];
    b1.y = bcol1[(kb + 1) * BPITCH];
    c00 = wmma_f32(a0, b0, c00);
    c01 = wmma_f32(a0, b1, c01);
    c10 = wmma_f32(a1, b0, c10);
    c11 = wmma_f32(a1, b1, c11);
  }

#pragma unroll
  for (int tj = 0; tj < 2; ++tj) {
    const int col = col0 + wn * 32 + tj * 16 + m;     // C/D layout: N = lane&15
    const float bv = bias[col];
#pragma unroll
    for (int ti = 0; ti < 2; ++ti) {
      const v8f& acc = (tj == 0) ? (ti == 0 ? c00 : c10) : (ti == 0 ? c01 : c11);
#pragma unroll
      for (int i = 0; i < 8; ++i) {
        const int r = mb + wm * 32 + ti * 16 + i + 8 * half;  // M = i + 8*half
        float val = acc[i] + bv;
        if (RELU) val = fmaxf(val, 0.f);
        if (RES)  val += R[(size_t)r * EMBED + col];
        out[(size_t)r * EMBED + col] = val;
      }
    }
  }
}

// ---------------------------------------------------------------------------
// Fused causal attention for one (b, h) and one 32-query block.
// Q/K/V stored as [B*T, C] with column offset h*64 (head-concat layout).
// Q (32x64), K (256x64), V (256x64) panels DMA'd into LDS by the TDM
// (Q/K pitch 68, V pitch 72), drained with s_wait_tensorcnt.
// Phase 1: S = Q K^T * scale (+mask) -> LDS, 2x2 tile reuse per wave (WMMA)
// Phase 2: row softmax in LDS (shfl reductions)
// Phase 3: O = P V                                                   (WMMA)
// ---------------------------------------------------------------------------
__global__ void attn_kernel(const float* __restrict__ Q,
                            const float* __restrict__ K,
                            const float* __restrict__ V,
                            float* __restrict__ O) {
  constexpr int SP = 260;                        // score pitch (260%64==4)
  __shared__ __align__(16) float S[32][SP];      // 33.3 KB
  extern __shared__ float dyn[];                 // Q | K | V panels (152 KB)
  float* Qs = dyn;                               // 32  x QKPITCH
  float* Ks = dyn + 32 * QKPITCH;                // 256 x QKPITCH
  float* Vs = Ks + SEQ * QKPITCH;                // 256 x VPITCH

  const int bh   = blockIdx.x;                   // 0..B*H-1
  const int b    = bh / NHEAD;
  const int h    = bh % NHEAD;
  const int hoff = h * HEADD;
  const int q0   = blockIdx.y * 32;              // query block start
  const int wave = threadIdx.x >> 5;
  const int lane = threadIdx.x & 31;
  const int m    = lane & 15;
  const int half = lane >> 4;
  const size_t base = (size_t)b * SEQ * EMBED + hoff;
  const float scale = 0.051031036307982884f;     // 1/sqrt(384)

  // ---- TDM: stage Q, K, V panels into LDS ---------------------------------
  if (wave == 0) {
    tdm_load_2d(Q + base + (size_t)q0 * EMBED, (unsigned)(uintptr_t)Qs,
                32, EMBED, 3 /*pad 4 -> pitch 68*/);
    tdm_load_2d(K + base, (unsigned)(uintptr_t)Ks, SEQ, EMBED, 3);
    tdm_load_2d(V + base, (unsigned)(uintptr_t)Vs, SEQ, EMBED, 7 /*pitch 72*/);
    __builtin_amdgcn_s_wait_tensorcnt(0);
  }
  __syncthreads();

  // ---- phase 1: 2x2 score tiles per wave (mt 0..1 x nt {w, w+8}) ----------
  {
    const int nt0 = wave, nt1 = wave + 8;
    const float* qr0 = Qs + (m)      * QKPITCH + 2 * half;   // mt=0
    const float* qr1 = Qs + (16 + m) * QKPITCH + 2 * half;   // mt=1
    const float* kr0 = Ks + (nt0 * 16 + m) * QKPITCH + 2 * half;
    const float* kr1 = Ks + (nt1 * 16 + m) * QKPITCH + 2 * half;
    v8f s00 = {}, s01 = {}, s10 = {}, s11 = {};
#pragma unroll 4
    for (int kb = 0; kb < HEADD; kb += 4) {
      const v2f a0 = *(const v2f*)(qr0 + kb);
      const v2f a1 = *(const v2f*)(qr1 + kb);
      const v2f b0 = *(const v2f*)(kr0 + kb);
      const v2f b1 = *(const v2f*)(kr1 + kb);
      s00 = wmma_f32(a0, b0, s00);
      s01 = wmma_f32(a0, b1, s01);
      s10 = wmma_f32(a1, b0, s10);
      s11 = wmma_f32(a1, b1, s11);
    }
#pragma unroll
    for (int tj = 0; tj < 2; ++tj) {
      const int nt  = tj ? nt1 : nt0;
      const int key = nt * 16 + m;               // key index (m acts as n)
#pragma unroll
      for (int ti = 0; ti < 2; ++ti) {
        const v8f& acc = (tj == 0) ? (ti == 0 ? s00 : s10) : (ti == 0 ? s01 : s11);
#pragma unroll
        for (int i = 0; i < 8; ++i) {
          const int mm = ti * 16 + i + 8 * half; // local query row
          S[mm][key] = (key <= q0 + mm) ? acc[i] * scale : -3.0e38f;
        }
      }
    }
  }
  __syncthreads();

  // ---- phase 2: softmax over 256 keys, 4 rows per wave --------------------
#pragma unroll
  for (int r = 0; r < 4; ++r) {
    const int mm = wave * 4 + r;
    float v[8];
    float mx = -3.0e38f;
#pragma unroll
    for (int j = 0; j < 8; ++j) { v[j] = S[mm][lane + 32 * j]; mx = fmaxf(mx, v[j]); }
#pragma unroll
    for (int off = 16; off > 0; off >>= 1) mx = fmaxf(mx, __shfl_xor(mx, off, 32));
    float sum = 0.f;
#pragma unroll
    for (int j = 0; j < 8; ++j) { v[j] = __expf(v[j] - mx); sum += v[j]; }
#pragma unroll
    for (int off = 16; off > 0; off >>= 1) sum += __shfl_xor(sum, off, 32);
    const float inv = 1.f / sum;
#pragma unroll
    for (int j = 0; j < 8; ++j) S[mm][lane + 32 * j] = v[j] * inv;
  }
  __syncthreads();

  // ---- phase 3: O = P @ V, one 16x16 output tile per wave (2 x 4) ---------
  {
    const int mt = wave >> 2;                    // M-tile (query rows)
    const int dt = wave & 3;                     // 16-wide slice of D=64
    const float* vb = Vs + (2 * half) * VPITCH + dt * 16 + m;  // m acts as n
    v8f acc = {};
#pragma unroll 2
    for (int kb = 0; kb < SEQ; kb += 4) {
      const v2f af = *(const v2f*)(&S[mt * 16 + m][kb + 2 * half]);  // A = P
      v2f bf;
      bf.x = vb[kb * VPITCH];
      bf.y = vb[(kb + 1) * VPITCH];
      acc = wmma_f32(af, bf, acc);
    }
#pragma unroll
    for (int i = 0; i < 8; ++i) {
      const int r = q0 + mt * 16 + i + 8 * half;
      O[base + (size_t)r * EMBED + dt * 16 + m] = acc[i];
    }
  }
}

// ---------------------------------------------------------------------------
// Host-side orchestration (all on `stream`, graph-capture safe).
// ---------------------------------------------------------------------------
extern "C" void kernel_launch(void* const* d_in, const int* in_sizes, int n_in,
                              void* d_out, int out_size, void* d_ws, size_t ws_size,
                              hipStream_t stream) {
  const float* x    = (const float*)d_in[0];
  const float* Wq   = (const float*)d_in[1];
  const float* bq   = (const float*)d_in[2];
  const float* Wk   = (const float*)d_in[3];
  const float* bk   = (const float*)d_in[4];
  const float* Wv   = (const float*)d_in[5];
  const float* bv   = (const float*)d_in[6];
  const float* Wp   = (const float*)d_in[7];
  const float* bp   = (const float*)d_in[8];
  const float* W1   = (const float*)d_in[9];
  const float* b1   = (const float*)d_in[10];
  const float* W2   = (const float*)d_in[11];
  const float* b2   = (const float*)d_in[12];
  const float* ln1g = (const float*)d_in[13];
  const float* ln1b = (const float*)d_in[14];
  const float* ln2g = (const float*)d_in[15];
  const float* ln2b = (const float*)d_in[16];
  float* out = (float*)d_out;

  const size_t MC = (size_t)MROWS * EMBED;       // 6,291,456 floats
  float* ws   = (float*)d_ws;                    // needs ~201 MB
  float* h    = ws + 0 * MC;
  float* q    = ws + 1 * MC;
  float* k    = ws + 2 * MC;
  float* v    = ws + 3 * MC;
  float* attn = ws + 4 * MC;
  float* x2   = ws + 5 * MC;
  float* h2   = ws + 6 * MC;
  float* ff1  = ws + 7 * MC;

  const dim3   gblk(EMBED / 64, MROWS / 128);    // (6, 128)
  const size_t gemm_lds = (size_t)EMBED * BPITCH * sizeof(float);  // 110592 B
  const size_t attn_lds =
      (size_t)(32 * QKPITCH + SEQ * QKPITCH + SEQ * VPITCH) * sizeof(float); // 152064 B
  constexpr int HBS = EMBED * HEADD;             // head-blocked weight stride

  // h = LN1(x)
  ln_kernel<<<MROWS / 8, 256, 0, stream>>>(x, ln1g, ln1b, h);
  // q,k,v = h @ W{q,k,v} + b  (head-blocked weights, concat-head layout)
  gemm_kernel<HBS, HEADD, false, false><<<gblk, 256, gemm_lds, stream>>>(h, Wq, bq, nullptr, q);
  gemm_kernel<HBS, HEADD, false, false><<<gblk, 256, gemm_lds, stream>>>(h, Wk, bk, nullptr, k);
  gemm_kernel<HBS, HEADD, false, false><<<gblk, 256, gemm_lds, stream>>>(h, Wv, bv, nullptr, v);
  // attn = softmax(mask(q k^T / sqrt(C))) v   (fused, per (b,h,32-row block))
  attn_kernel<<<dim3(BATCH * NHEAD, SEQ / 32), 256, attn_lds, stream>>>(q, k, v, attn);
  // x2 = x + attn @ Wp + bp
  gemm_kernel<64, EMBED, false, true><<<gblk, 256, gemm_lds, stream>>>(attn, Wp, bp, x, x2);
  // h2 = LN2(x2)
  ln_kernel<<<MROWS / 8, 256, 0, stream>>>(x2, ln2g, ln2b, h2);
  // ff1 = relu(h2 @ W1 + b1)
  gemm_kernel<64, EMBED, true, false><<<gblk, 256, gemm_lds, stream>>>(h2, W1, b1, nullptr, ff1);
  // out = x2 + ff1 @ W2 + b2
  gemm_kernel<64, EMBED, false, true><<<gblk, 256, gemm_lds, stream>>>(ff1, W2, b2, x2, out);
}